// GIN_28879360098945
// MI455X (gfx1250) — compile-verified
//
#include <hip/hip_runtime.h>

#define N_NODES 50000
#define N_EDGES 800000
#define DIM     128
#define N_LAYERS 3
#define BN_EPS  1e-5f

typedef __attribute__((ext_vector_type(2))) float v2f;
typedef __attribute__((ext_vector_type(8))) float v8f;

// -------------------- scatter: agg[dst] += relu(h[src]) --------------------
// One wave32 per edge; lane handles 4 consecutive features (32*4 = 128 = DIM).
__global__ void gin_scatter_kernel(const float* __restrict__ h,
                                   const int* __restrict__ ei,
                                   float* __restrict__ agg) {
    int wave = (blockIdx.x * blockDim.x + threadIdx.x) >> 5;
    int lane = threadIdx.x & 31;
    if (wave >= N_EDGES) return;
    int src = ei[wave];
    int dst = ei[N_EDGES + wave];
    float4 v = ((const float4*)(h + (size_t)src * DIM))[lane];
    v.x = fmaxf(v.x, 0.0f);
    v.y = fmaxf(v.y, 0.0f);
    v.z = fmaxf(v.z, 0.0f);
    v.w = fmaxf(v.w, 0.0f);
    float* ap = agg + (size_t)dst * DIM + lane * 4;
    atomicAdd(ap + 0, v.x);
    atomicAdd(ap + 1, v.y);
    atomicAdd(ap + 2, v.z);
    atomicAdd(ap + 3, v.w);
}

// -------------------- z = (1+eps)*h + agg --------------------
__global__ void gin_axpy_kernel(const float* __restrict__ h,
                                const float* __restrict__ agg,
                                const float* __restrict__ epsp, int layer,
                                float* __restrict__ z) {
    int i = blockIdx.x * blockDim.x + threadIdx.x;     // float4 index
    if (i >= (N_NODES * DIM) / 4) return;
    float s = 1.0f + epsp[layer];
    float4 hv = ((const float4*)h)[i];
    float4 av = ((const float4*)agg)[i];
    float4 r;
    r.x = s * hv.x + av.x;
    r.y = s * hv.y + av.y;
    r.z = s * hv.z + av.z;
    r.w = s * hv.w + av.w;
    ((float4*)z)[i] = r;
}

// -------------------- Y = A(nrows x 128) * W(128 x 128) + bias --------------------
// Block = 256 threads = 8 waves. Block owns 16 rows; wave w owns cols [16w,16w+16).
// A tile and transposed W staged in LDS with pad stride 132 (bank-conflict free).
__global__ __launch_bounds__(256)
void gin_gemm_wmma_kernel(const float* __restrict__ A,
                          const float* __restrict__ W,
                          const float* __restrict__ bias,
                          float* __restrict__ Y) {
    __shared__ float At[16][132];    // At[row][k]
    __shared__ float Wt[128][132];   // Wt[col][k]  (transposed weight)
    const int t       = threadIdx.x;
    const int lane    = t & 31;
    const int waveId  = t >> 5;
    const int rowBase = blockIdx.x * 16;

    // Stage A tile: 16 rows x 128 cols = 512 float4, 2 per thread, coalesced.
#pragma unroll
    for (int it = 0; it < 2; ++it) {
        int q  = it * 256 + t;     // 0..511
        int r  = q >> 5;           // 0..15
        int c4 = q & 31;           // 0..31
        float4 v = ((const float4*)(A + (size_t)(rowBase + r) * DIM))[c4];
        *(float4*)&At[r][c4 * 4] = v;
    }
    // Stage W transposed: task q -> kquad kq (0..31), col c (0..127).
    // Lanes sweep consecutive c: global reads coalesced, LDS b128 stores ~conflict-free.
#pragma unroll
    for (int it = 0; it < 16; ++it) {
        int q  = it * 256 + t;     // 0..4095
        int kq = q >> 7;           // 0..31
        int c  = q & 127;
        float4 v;
        v.x = W[(kq * 4 + 0) * DIM + c];
        v.y = W[(kq * 4 + 1) * DIM + c];
        v.z = W[(kq * 4 + 2) * DIM + c];
        v.w = W[(kq * 4 + 3) * DIM + c];
        *(float4*)&Wt[c][kq * 4] = v;
    }
    __syncthreads();

    const int half    = lane >> 4;   // K-pair select per ISA A/B layout
    const int lm      = lane & 15;   // M index (A) / N index (B)
    const int colBase = waveId * 16;

    v8f acc = {};
#pragma unroll
    for (int kk = 0; kk < 128; kk += 4) {
        int k = kk + 2 * half;
        v2f a = *(const v2f*)&At[lm][k];              // A[lm][k], A[lm][k+1]
        v2f b = *(const v2f*)&Wt[colBase + lm][k];    // W[k][col], W[k+1][col]
        acc = __builtin_amdgcn_wmma_f32_16x16x4_f32(
            /*neg_a=*/false, a, /*neg_b=*/false, b,
            /*c_mod=*/(short)0, acc, /*reuse_a=*/false, /*reuse_b=*/false);
    }

    // C/D layout: VGPR r -> row (r + 8*half), col = colBase + lm.
    const int col = colBase + lm;
    const float bv = bias[col];
    float* yp = Y + (size_t)(rowBase + 8 * half) * DIM + col;
#pragma unroll
    for (int r = 0; r < 8; ++r)
        yp[(size_t)r * DIM] = acc[r] + bv;
}

// -------------------- per-column sum / sumsq --------------------
__global__ void gin_colstats_kernel(const float* __restrict__ Y,
                                    float* __restrict__ stats) {
    __shared__ float sbuf[256];
    __shared__ float qbuf[256];
    int c  = threadIdx.x & 127;
    int r0 = blockIdx.x * 2 + (threadIdx.x >> 7);
    float s = 0.0f, q = 0.0f;
    for (int r = r0; r < N_NODES; r += gridDim.x * 2) {
        float v = Y[(size_t)r * DIM + c];
        s += v;
        q += v * v;
    }
    sbuf[threadIdx.x] = s;
    qbuf[threadIdx.x] = q;
    __syncthreads();
    if (threadIdx.x < 128) {
        s = sbuf[threadIdx.x] + sbuf[threadIdx.x + 128];
        q = qbuf[threadIdx.x] + qbuf[threadIdx.x + 128];
        atomicAdd(&stats[c], s);
        atomicAdd(&stats[DIM + c], q);
    }
}

// -------------------- BN (training-mode) + optional ReLU --------------------
__global__ void gin_bn_kernel(const float* __restrict__ Y,
                              const float* __restrict__ stats,
                              const float* __restrict__ g,
                              const float* __restrict__ b,
                              float* __restrict__ out, int do_relu) {
    int i = blockIdx.x * blockDim.x + threadIdx.x;     // float4 index
    if (i >= (N_NODES * DIM) / 4) return;
    int c0 = (i & 31) * 4;
    float4 y = ((const float4*)Y)[i];
    float4 o;
    const float inv_n = 1.0f / (float)N_NODES;
#pragma unroll
    for (int j = 0; j < 4; ++j) {
        int c = c0 + j;
        float mu   = stats[c] * inv_n;
        float var  = stats[DIM + c] * inv_n - mu * mu;   // biased, matches jnp.var
        float rstd = rsqrtf(var + BN_EPS);
        float v = ((&y.x)[j] - mu) * rstd * g[c] + b[c];
        if (do_relu) v = fmaxf(v, 0.0f);
        (&o.x)[j] = v;
    }
    ((float4*)out)[i] = o;
}

extern "C" void kernel_launch(void* const* d_in, const int* in_sizes, int n_in,
                              void* d_out, int out_size, void* d_ws, size_t ws_size,
                              hipStream_t stream) {
    const float* x    = (const float*)d_in[0];
    const int*   ei   = (const int*)d_in[1];     // (2, E) int
    const float* eps  = (const float*)d_in[2];   // (L,)
    const float* W1   = (const float*)d_in[3];   // (L, D, D)
    const float* b1   = (const float*)d_in[4];
    const float* bn1g = (const float*)d_in[5];
    const float* bn1b = (const float*)d_in[6];
    const float* W2   = (const float*)d_in[7];
    const float* b2   = (const float*)d_in[8];
    const float* bn2g = (const float*)d_in[9];
    const float* bn2b = (const float*)d_in[10];

    const size_t ND = (size_t)N_NODES * DIM;
    float* bufA  = (float*)d_ws;        // agg, then GEMM output y
    float* bufZ  = bufA + ND;           // z, then relu'd intermediate
    float* bufH  = bufZ + ND;           // layer output h
    float* stats = bufH + ND;           // 2*DIM

    const int ew_blocks = (N_EDGES * 32) / 256;      // 1 wave per edge
    const int v4_blocks = (int)(ND / 4 / 256);       // float4 elementwise
    const int gm_blocks = N_NODES / 16;              // 16-row GEMM tiles

    const float* hin = x;
    for (int l = 0; l < N_LAYERS; ++l) {
        const size_t wo = (size_t)l * DIM * DIM;
        const size_t vo = (size_t)l * DIM;

        hipMemsetAsync(bufA, 0, ND * sizeof(float), stream);
        gin_scatter_kernel<<<ew_blocks, 256, 0, stream>>>(hin, ei, bufA);
        gin_axpy_kernel<<<v4_blocks, 256, 0, stream>>>(hin, bufA, eps, l, bufZ);

        // y = z @ W1 + b1  -> bufA ; BN1 + ReLU -> bufZ
        gin_gemm_wmma_kernel<<<gm_blocks, 256, 0, stream>>>(bufZ, W1 + wo, b1 + vo, bufA);
        hipMemsetAsync(stats, 0, 2 * DIM * sizeof(float), stream);
        gin_colstats_kernel<<<256, 256, 0, stream>>>(bufA, stats);
        gin_bn_kernel<<<v4_blocks, 256, 0, stream>>>(bufA, stats, bn1g + vo, bn1b + vo, bufZ, 1);

        // y2 = r @ W2 + b2 -> bufA ; BN2 (+ReLU unless last) -> h
        gin_gemm_wmma_kernel<<<gm_blocks, 256, 0, stream>>>(bufZ, W2 + wo, b2 + vo, bufA);
        hipMemsetAsync(stats, 0, 2 * DIM * sizeof(float), stream);
        gin_colstats_kernel<<<256, 256, 0, stream>>>(bufA, stats);
        float* outp = (l == N_LAYERS - 1) ? (float*)d_out : bufH;
        gin_bn_kernel<<<v4_blocks, 256, 0, stream>>>(bufA, stats, bn2g + vo, bn2b + vo,
                                                     outp, (l < N_LAYERS - 1) ? 1 : 0);
        hin = bufH;
    }
}